// FeatureNormSingle_60825326846431
// MI455X (gfx1250) — compile-verified
//
#include <hip/hip_runtime.h>
#include <math.h>
#include <stdint.h>

// Problem shape (fixed by the reference setup).
#define B_  64
#define T_  1000
#define F_  257
#define C_  8            // time chunks for the parallel (chunked) scan
#define L_  (T_ / C_)    // 125 steps per chunk
#define NB  4            // b-chains carried per thread (ILP)
#define FT  ((F_ + 31) / 32)   // 9 f-tiles of 32 lanes
#define BF  (B_ * F_)          // 16448 chains
#define EPS 1e-8f
#define PFD 32           // prefetch distance in time-rows

// Pass-3 TDM double-buffer geometry.
#define TSTEP  5                 // time-rows per LDS tile (125 = 25 * 5)
#define NSTEPS (L_ / TSTEP)      // 25 pipeline steps
#define BLOCK3 128               // 4 waves per block
#define WAVES3 (BLOCK3 / 32)
#define WAVE_TILE_FLTS (2 * NB * TSTEP * 32)   // double buffer per wave (5 KB)

// Toolchain arity detection for the TDM builtin (5-arg ROCm vs 6-arg therock).
#if __has_include(<hip/amd_detail/amd_gfx1250_TDM.h>)
#define TDM_ARGS6 1
#else
#define TDM_ARGS6 0
#endif

typedef unsigned int u32x4 __attribute__((ext_vector_type(4)));
typedef int          i32x4 __attribute__((ext_vector_type(4)));
typedef int          i32x8 __attribute__((ext_vector_type(8)));

// LDS byte offset of a generic pointer into __shared__ (AS3 ptrtoint).
__device__ __forceinline__ uint32_t lds_offset_of(void* p) {
    return (uint32_t)(uintptr_t)(__attribute__((address_space(3))) void*)p;
}

// One TDM descriptor: 3D tile, x = tile_dim0 elems (f), y = TSTEP rows
// (stride F), z = NB streams (stride T*F). Lands contiguously in LDS.
__device__ __forceinline__ void tdm_load_tile(uint64_t gaddr, uint32_t lds_addr,
                                              i32x8 g1, i32x4 g2, i32x4 g3) {
    u32x4 g0;
    g0[0] = 1u;                                              // count=1, user D#
    g0[1] = lds_addr;                                        // lds_addr (bytes)
    g0[2] = (uint32_t)(gaddr & 0xFFFFFFFFull);               // global_addr[31:0]
    g0[3] = (uint32_t)((gaddr >> 32) & 0x1FFFFFFull)         // global_addr[56:32]
          | (2u << 30);                                      // type = 2 ("image")
#if TDM_ARGS6
    i32x8 g4 = {0, 0, 0, 0, 0, 0, 0, 0};
    __builtin_amdgcn_tensor_load_to_lds(g0, g1, g2, g3, g4, 0);
#else
    __builtin_amdgcn_tensor_load_to_lds(g0, g1, g2, g3, 0);
#endif
}

template <int N>
__device__ __forceinline__ void wait_tensorcnt() {
    __builtin_amdgcn_s_wait_tensorcnt(N);
    // TDM's LDS writes are invisible to alias analysis (address lives inside
    // an integer descriptor) -- fence the compiler before consuming the tile.
    asm volatile("" ::: "memory");
}

__device__ __forceinline__ float sigmoid_fast(float x) {
    return 1.0f / (1.0f + __expf(-x));
}

// ---------------------------------------------------------------------------
// Pass 1: per-chunk local scan with zero carry-in; emit chunk-end value.
// wave -> (b-group of 4, f-tile of 32, chunk). Lane l handles f = tile*32+l.
// Direct global loads + deep global_prefetch_b8 pipelining (HBM-bound pass).
// chunk_end layout: [c][b][f]
// ---------------------------------------------------------------------------
__global__ __launch_bounds__(256) void ema_local_scan(
    const float* __restrict__ mag,
    const float* __restrict__ alpha,
    float* __restrict__ chunk_end) {
    const int tid   = blockIdx.x * blockDim.x + threadIdx.x;
    const int lane  = tid & 31;
    const int wave  = tid >> 5;
    const int c     = wave % C_;
    const int tmp   = wave / C_;
    const int ftile = tmp % FT;
    const int bg    = tmp / FT;           // 0..15
    const int f     = ftile * 32 + lane;
    const bool valid = (f < F_);
    const int fc    = valid ? f : (F_ - 1);   // clamp so loads stay in-bounds

    const float a   = sigmoid_fast(alpha[fc]);
    const float oma = 1.0f - a;
    const int  b0   = bg * NB;

    const size_t tbase = (size_t)c * L_;
    const float* p0 = mag + ((size_t)(b0 + 0) * T_ + tbase) * F_ + fc;
    const float* p1 = mag + ((size_t)(b0 + 1) * T_ + tbase) * F_ + fc;
    const float* p2 = mag + ((size_t)(b0 + 2) * T_ + tbase) * F_ + fc;
    const float* p3 = mag + ((size_t)(b0 + 3) * T_ + tbase) * F_ + fc;

    float s0 = 0.0f, s1 = 0.0f, s2 = 0.0f, s3 = 0.0f;

#pragma unroll 5
    for (int t = 0; t < L_; ++t) {
        const size_t off = (size_t)t * F_;
        // Deep prefetch (global_prefetch_b8): speculative, OOB is dropped,
        // and it does not consume LOADcnt entries.
        __builtin_prefetch(p0 + off + (size_t)PFD * F_, 0, 0);
        __builtin_prefetch(p1 + off + (size_t)PFD * F_, 0, 0);
        __builtin_prefetch(p2 + off + (size_t)PFD * F_, 0, 0);
        __builtin_prefetch(p3 + off + (size_t)PFD * F_, 0, 0);
        const float m0 = p0[off];
        const float m1 = p1[off];
        const float m2 = p2[off];
        const float m3 = p3[off];
        s0 = fmaf(oma, s0, a * (m0 * m0));
        s1 = fmaf(oma, s1, a * (m1 * m1));
        s2 = fmaf(oma, s2, a * (m2 * m2));
        s3 = fmaf(oma, s3, a * (m3 * m3));
    }

    if (valid) {
        float* dst = chunk_end + (size_t)c * BF + f;
        dst[(size_t)(b0 + 0) * F_] = s0;
        dst[(size_t)(b0 + 1) * F_] = s1;
        dst[(size_t)(b0 + 2) * F_] = s2;
        dst[(size_t)(b0 + 3) * F_] = s3;
    }
}

// ---------------------------------------------------------------------------
// Pass 2 (tiny): propagate carries across chunks (EMA is linear):
// carry[c] = (1-a)^L * carry[c-1] + local_end[c-1], carry[0] = s_init.
// ---------------------------------------------------------------------------
__global__ __launch_bounds__(256) void ema_combine(
    const float* __restrict__ s_init,
    const float* __restrict__ alpha,
    const float* __restrict__ chunk_end,
    float* __restrict__ carry_in) {
    const int idx = blockIdx.x * blockDim.x + threadIdx.x;  // b*F + f
    if (idx >= BF) return;
    const int f = idx % F_;
    const float a      = sigmoid_fast(alpha[f]);
    const float oma    = 1.0f - a;
    const float decayL = __powf(oma, (float)L_);

    float carry = s_init[idx];
#pragma unroll
    for (int c = 0; c < C_; ++c) {
        carry_in[(size_t)c * BF + idx] = carry;
        carry = fmaf(decayL, carry, chunk_end[(size_t)c * BF + idx]);
    }
}

// ---------------------------------------------------------------------------
// Pass 3: rerun each chunk's scan with the correct carry-in and write output:
// out = mag * rcp(sqrt(s)+eps) * w + bias.
// mag is re-read out of the 192 MB global L2 (warmed by pass 1). Each wave
// double-buffers its [NB x TSTEP x 32] tile in LDS via ONE Tensor Data Mover
// descriptor per fill (TENSORcnt-tracked, 2-deep pipeline) and consumes it
// with ds_load_b32 while the next DMA is in flight.
// ---------------------------------------------------------------------------
__global__ __launch_bounds__(BLOCK3) void ema_final(
    const float* __restrict__ mag,
    const float* __restrict__ weights,
    const float* __restrict__ bias,
    const float* __restrict__ alpha,
    const float* __restrict__ carry_in,
    float* __restrict__ out) {
    __shared__ float smem[WAVES3 * WAVE_TILE_FLTS];

    const int tid   = blockIdx.x * BLOCK3 + threadIdx.x;
    const int lane  = tid & 31;
    const int wave  = tid >> 5;
    const int c     = wave % C_;
    const int tmp   = wave / C_;
    const int ftile = tmp % FT;
    const int bg    = tmp / FT;
    const int f     = ftile * 32 + lane;
    const bool valid = (f < F_);
    const int fc    = valid ? f : (F_ - 1);

    const float a   = sigmoid_fast(alpha[fc]);
    const float oma = 1.0f - a;
    const float w   = weights[fc];
    const float bi  = bias[fc];
    const int  b0   = bg * NB;

    const size_t tbase = (size_t)c * L_;
    float* q0 = out + ((size_t)(b0 + 0) * T_ + tbase) * F_ + fc;
    float* q1 = out + ((size_t)(b0 + 1) * T_ + tbase) * F_ + fc;
    float* q2 = out + ((size_t)(b0 + 2) * T_ + tbase) * F_ + fc;
    float* q3 = out + ((size_t)(b0 + 3) * T_ + tbase) * F_ + fc;

    const float* ci = carry_in + (size_t)c * BF + fc;
    float s0 = ci[(size_t)(b0 + 0) * F_];
    float s1 = ci[(size_t)(b0 + 1) * F_];
    float s2 = ci[(size_t)(b0 + 2) * F_];
    float s3 = ci[(size_t)(b0 + 3) * F_];

    float* sw = smem + (threadIdx.x >> 5) * WAVE_TILE_FLTS;
    const uint32_t sw_lds  = lds_offset_of(sw);
    // Wave-uniform global byte address of the tile origin (f = ftile*32).
    const uint64_t g_base  = (uint64_t)(uintptr_t)mag
                           + 4ull * (((uint64_t)b0 * T_ + tbase) * F_ + (uint64_t)ftile * 32);

    // D# group 1: data_size=4B; tensor_dim0=rem (OOB f reads return 0),
    // tensor_dim1=TSTEP, tile_dim0=32, tile_dim1=TSTEP, tile_dim2=NB;
    // dim0_stride=F, dim1_stride=T*F (z steps across b-streams).
    const uint32_t rem = (uint32_t)(F_ - ftile * 32) > 32u ? 32u
                       : (uint32_t)(F_ - ftile * 32);
    const uint32_t zstride = (uint32_t)(T_ * F_);            // 257000 < 2^48
    i32x8 g1;
    g1[0] = (int)(2u << 16);                 // workgroup_mask=0, data_size=2 (4B)
    g1[1] = (int)((rem & 0xFFFFu) << 16);    // tensor_dim0[15:0] @ bits 63:48
    g1[2] = (int)(((uint32_t)TSTEP) << 16);  // dim0 hi=0 | tensor_dim1[15:0]
    g1[3] = (int)(32u << 16);                // dim1 hi=0 | tile_dim0=32
    g1[4] = (int)TSTEP;                      // tile_dim1=TSTEP | tile_dim2=0 (set below)
    g1[5] = (int)F_;                         // tensor_dim0_stride low32 = 257
    g1[6] = (int)((zstride & 0xFFFFu) << 16);// dim0_stride hi=0 | dim1_stride lo16
    g1[7] = (int)(zstride >> 16);            // dim1_stride hi
    // Actually tile_dim2 lives in group1 bits 159:144:
    g1[4] = (int)(((uint32_t)NB << 16) | (uint32_t)TSTEP);   // tile_dim1 | tile_dim2=NB
    i32x4 g2 = {NB, 0, 0, 0};                // tensor_dim2 = NB bounds z
    i32x4 g3 = {0, 0, 0, 0};

    auto fill = [&](int buf, int t0) {
        tdm_load_tile(g_base + 4ull * (uint64_t)t0 * F_,
                      sw_lds + (uint32_t)buf * (NB * TSTEP * 32 * 4),
                      g1, g2, g3);
    };
    auto process = [&](int buf, int t0) {
#pragma unroll
        for (int r = 0; r < TSTEP; ++r) {
            const size_t off = (size_t)(t0 + r) * F_;
            const float m0 = sw[((buf * NB + 0) * TSTEP + r) * 32 + lane];
            const float m1 = sw[((buf * NB + 1) * TSTEP + r) * 32 + lane];
            const float m2 = sw[((buf * NB + 2) * TSTEP + r) * 32 + lane];
            const float m3 = sw[((buf * NB + 3) * TSTEP + r) * 32 + lane];
            s0 = fmaf(oma, s0, a * (m0 * m0));
            s1 = fmaf(oma, s1, a * (m1 * m1));
            s2 = fmaf(oma, s2, a * (m2 * m2));
            s3 = fmaf(oma, s3, a * (m3 * m3));
            const float i0 = __builtin_amdgcn_rcpf(__builtin_amdgcn_sqrtf(s0) + EPS);
            const float i1 = __builtin_amdgcn_rcpf(__builtin_amdgcn_sqrtf(s1) + EPS);
            const float i2 = __builtin_amdgcn_rcpf(__builtin_amdgcn_sqrtf(s2) + EPS);
            const float i3 = __builtin_amdgcn_rcpf(__builtin_amdgcn_sqrtf(s3) + EPS);
            if (valid) {
                q0[off] = fmaf(m0 * i0, w, bi);
                q1[off] = fmaf(m1 * i1, w, bi);
                q2[off] = fmaf(m2 * i2, w, bi);
                q3[off] = fmaf(m3 * i3, w, bi);
            }
        }
    };

    // 2-deep TDM pipeline: before consuming buffer X at most one newer
    // descriptor is outstanding (tensor ops complete in order per wave), so
    // s_wait_tensorcnt 1 guarantees X's tile landed. Drain with 0.
    fill(0, 0);
    fill(1, TSTEP);
    for (int step = 0; step < NSTEPS - 1; ++step) {
        wait_tensorcnt<1>();
        const int buf = step & 1;
        process(buf, step * TSTEP);
        if (step + 2 < NSTEPS) fill(buf, (step + 2) * TSTEP);
    }
    wait_tensorcnt<0>();
    process((NSTEPS - 1) & 1, (NSTEPS - 1) * TSTEP);
}

// ---------------------------------------------------------------------------
// Host launcher. Workspace: chunk_end [C][B][F] + carry_in [C][B][F] floats.
// ---------------------------------------------------------------------------
extern "C" void kernel_launch(void* const* d_in, const int* in_sizes, int n_in,
                              void* d_out, int out_size, void* d_ws, size_t ws_size,
                              hipStream_t stream) {
    const float* mag     = (const float*)d_in[0];
    const float* s_init  = (const float*)d_in[1];
    const float* weights = (const float*)d_in[2];
    const float* bias    = (const float*)d_in[3];
    const float* alpha   = (const float*)d_in[4];
    float* out = (float*)d_out;

    float* chunk_end = (float*)d_ws;
    float* carry_in  = chunk_end + (size_t)C_ * BF;

    // (B/NB) * FT * C waves of 32 lanes = 16*9*8*32 = 36864 threads
    const int nthreads = (B_ / NB) * FT * C_ * 32;

    ema_local_scan<<<nthreads / 256, 256, 0, stream>>>(mag, alpha, chunk_end);
    ema_combine<<<(BF + 255) / 256, 256, 0, stream>>>(s_init, alpha, chunk_end, carry_in);
    ema_final<<<nthreads / BLOCK3, BLOCK3, 0, stream>>>(mag, weights, bias, alpha,
                                                        carry_in, out);
}